// HgnnEncoder_17394617548830
// MI455X (gfx1250) — compile-verified
//
#include <hip/hip_runtime.h>
#include <math.h>

#define F_IN  128
#define D1    256
#define D2    64
#define H_ATT 128
#define L_HYE 16

typedef float v2f __attribute__((ext_vector_type(2)));
typedef float v8f __attribute__((ext_vector_type(8)));

struct Ptr4 { const float* p0; const float* p1; const float* p2; const float* p3; };

// ---------------------------------------------------------------------------
// C[M,Dout] = A[M,K] @ W[K,Dout] via V_WMMA_F32_16X16X4_F32.
// Requirements: M%16==0, K%4==0, Dout%(16*WAVES)==0 so EXEC is always all-1s.
// One wave owns one 16x16 output tile.
// A frag (16x4 f32): lane<16 -> K=k0+{0,1}, lane>=16 -> K=k0+{2,3}, row = lane&15
// B frag (4x16 f32): symmetric over columns.
// C/D (16x16 f32):  vgpr r, lane<16 -> M=r, lane>=16 -> M=r+8, N = lane&15
// ---------------------------------------------------------------------------
template<int WAVES>
__global__ __launch_bounds__(WAVES * 32)
void gemm_wmma_f32(const float* __restrict__ A, const float* __restrict__ W,
                   float* __restrict__ C, int M, int K, int Dout) {
  const int lane = threadIdx.x & 31;
  const int wave = threadIdx.x >> 5;
  const int tm = blockIdx.x;
  const int tn = blockIdx.y * WAVES + wave;
  const int mr = (tm << 4) + (lane & 15);
  const int nc = (tn << 4) + (lane & 15);
  const int kh = (lane >> 4) << 1;            // 0 or 2
  const float* __restrict__ arow = A + (size_t)mr * K + kh;
  const float* __restrict__ wcol = W + nc;
  v8f acc = {0.f, 0.f, 0.f, 0.f, 0.f, 0.f, 0.f, 0.f};
  for (int k0 = 0; k0 < K; k0 += 4) {
    const float2 av = *(const float2*)(arow + k0);   // 8B aligned: k0+kh even
    v2f a; a.x = av.x; a.y = av.y;
    v2f b;
    b.x = wcol[(size_t)(k0 + kh) * Dout];
    b.y = wcol[(size_t)(k0 + kh + 1) * Dout];
    acc = __builtin_amdgcn_wmma_f32_16x16x4_f32(false, a, false, b,
                                                (short)0, acc, false, false);
  }
  const int rbase = (tm << 4) + ((lane >> 4) << 3);
  float* __restrict__ crow = C + (size_t)rbase * Dout + nc;
#pragma unroll
  for (int r = 0; r < 8; ++r) crow[(size_t)r * Dout] = acc[r];
}

// ---------------------------------------------------------------------------
__global__ void deg_count_kernel(const int* __restrict__ nidx, const int* __restrict__ eidx,
                                 float* __restrict__ dn, float* __restrict__ de, int E) {
  int t = blockIdx.x * blockDim.x + threadIdx.x;
  if (t < E) {
    atomicAdd(&dn[nidx[t]], 1.0f);
    atomicAdd(&de[eidx[t]], 1.0f);
  }
}

__global__ void invert_kernel(float* __restrict__ d, int n) {
  int t = blockIdx.x * blockDim.x + threadIdx.x;
  if (t < n) { float v = d[t]; d[t] = v > 0.0f ? 1.0f / v : 0.0f; }
}

// dst[didx[p], :] += src[sidx[p], :] * (sscale ? sscale[sidx[p]] : 1)
template<int D>
__global__ void scatter_add_kernel(const float* __restrict__ src, const int* __restrict__ sidx,
                                   const int* __restrict__ didx, const float* __restrict__ sscale,
                                   float* __restrict__ dst, int E) {
  constexpr int CH = D / 4;
  long long t = blockIdx.x * (long long)blockDim.x + threadIdx.x;
  if (t >= (long long)E * CH) return;
  int pair = (int)(t / CH);
  int c    = (int)(t % CH);
  int srow = sidx[pair];
  int drow = didx[pair];
  const float4 v = *(const float4*)(src + (size_t)srow * D + c * 4);
  float sc = sscale ? sscale[srow] : 1.0f;
  float* p = dst + (size_t)drow * D + c * 4;
  atomicAdd(p + 0, v.x * sc);
  atomicAdd(p + 1, v.y * sc);
  atomicAdd(p + 2, v.z * sc);
  atomicAdd(p + 3, v.w * sc);
}

// out[r,d] = relu(in[r,d] * rscale[r] + bias[d])   (in-place safe)
template<int D>
__global__ void scale_bias_relu_kernel(const float* __restrict__ in, const float* __restrict__ rscale,
                                       const float* __restrict__ bias, float* __restrict__ out,
                                       long long total) {
  long long t = blockIdx.x * (long long)blockDim.x + threadIdx.x;
  if (t >= total) return;
  int row = (int)(t / D);
  int d   = (int)(t % D);
  float v = in[t] * rscale[row] + bias[d];
  out[t] = v > 0.0f ? v : 0.0f;
}

// out[n,:] = (sum_l h[hnode[n,l]-1, :] if hnode>0) / (hlen[n]+eps)
__global__ void hye_emb_kernel(const float* __restrict__ h, const int* __restrict__ hnode,
                               const float* __restrict__ hlen, float* __restrict__ out, int N) {
  int node = blockIdx.x * 4 + (threadIdx.x >> 6);
  int d = threadIdx.x & 63;
  if (node >= N) return;
  const int* hn = hnode + (size_t)node * L_HYE;
  float s = 0.0f;
#pragma unroll
  for (int l = 0; l < L_HYE; ++l) {
    int idx = hn[l];
    if (idx > 0) s += h[(size_t)(idx - 1) * D2 + d];
  }
  out[(size_t)node * D2 + d] = s / (hlen[node] + 1e-15f);
}

// scores[ch] += sum_rows tanh(z_row @ W1 + b1) @ W2
__global__ void attn_score_kernel(Ptr4 zs, const float* __restrict__ W1,
                                  const float* __restrict__ b1, const float* __restrict__ W2,
                                  float* __restrict__ scores, int N) {
  const float* z = (blockIdx.y == 0) ? zs.p0 : (blockIdx.y == 1) ? zs.p1
                 : (blockIdx.y == 2) ? zs.p2 : zs.p3;
  int row = blockIdx.x * blockDim.x + threadIdx.x;
  float val = 0.0f;
  if (row < N) {
    float4 zz[D2 / 4];
    const float4* zp = (const float4*)(z + (size_t)row * D2);
#pragma unroll
    for (int j = 0; j < D2 / 4; ++j) zz[j] = zp[j];
    for (int h = 0; h < H_ATT; ++h) {
      float s = b1[h];
#pragma unroll 4
      for (int j = 0; j < D2 / 4; ++j) {
        const float* wc = W1 + (size_t)(4 * j) * H_ATT + h;   // wave-uniform loads
        s += zz[j].x * wc[0] + zz[j].y * wc[H_ATT]
           + zz[j].z * wc[2 * H_ATT] + zz[j].w * wc[3 * H_ATT];
      }
      val += tanhf(s) * W2[h];
    }
  }
  __shared__ float red[256];
  red[threadIdx.x] = val;
  __syncthreads();
  for (int s = 128; s > 0; s >>= 1) {
    if ((int)threadIdx.x < s) red[threadIdx.x] += red[threadIdx.x + s];
    __syncthreads();
  }
  if (threadIdx.x == 0) atomicAdd(&scores[blockIdx.y], red[0]);
}

__global__ void softmax_beta_kernel(const float* __restrict__ scores,
                                    float* __restrict__ beta, float invN) {
  for (int hd = 0; hd < 2; ++hd) {
    float w[4]; float mx = -1e30f;
    for (int k = 0; k < 4; ++k) { w[k] = scores[hd * 4 + k] * invN; mx = fmaxf(mx, w[k]); }
    float s = 0.0f;
    for (int k = 0; k < 4; ++k) { w[k] = __expf(w[k] - mx); s += w[k]; }
    for (int k = 0; k < 4; ++k) beta[hd * 4 + k] = w[k] / s;
  }
}

__global__ void combine_kernel(Ptr4 zs, const float* __restrict__ beta,
                               float* __restrict__ out, long long total) {
  long long t = blockIdx.x * (long long)blockDim.x + threadIdx.x;
  if (t >= total) return;
  out[t] = beta[0] * zs.p0[t] + beta[1] * zs.p1[t]
         + beta[2] * zs.p2[t] + beta[3] * zs.p3[t];
}

// ---------------------------------------------------------------------------
extern "C" void kernel_launch(void* const* d_in, const int* in_sizes, int n_in,
                              void* d_out, int out_size, void* d_ws, size_t ws_size,
                              hipStream_t stream) {
  (void)n_in; (void)out_size; (void)ws_size;
  const int N = in_sizes[0] / F_IN;
  const int E = in_sizes[1] / 2;

  const float* W1   = (const float*)d_in[16];
  const float* b1   = (const float*)d_in[17];
  const float* W2   = (const float*)d_in[18];
  const float* b2   = (const float*)d_in[19];
  const float* acW1 = (const float*)d_in[20];
  const float* acb1 = (const float*)d_in[21];
  const float* acW2 = (const float*)d_in[22];
  const float* amW1 = (const float*)d_in[23];
  const float* amb1 = (const float*)d_in[24];
  const float* amW2 = (const float*)d_in[25];

  float* ws = (float*)d_ws;
  size_t off = 0;
  auto alloc = [&](size_t nfloats) {
    float* p = ws + off;
    off += (nfloats + 63) & ~(size_t)63;
    return p;
  };
  float* h2[4]; float* he[4];
  for (int g = 0; g < 4; ++g) h2[g] = alloc((size_t)N * D2);
  for (int g = 0; g < 4; ++g) he[g] = alloc((size_t)N * D2);
  float* buf0   = alloc((size_t)N * D1);
  float* buf1   = alloc((size_t)N * D1);
  float* degn   = alloc((size_t)N);
  float* dege   = alloc((size_t)N);
  float* scores = alloc(8);
  float* beta   = alloc(8);

  const int TB = 256;
  dim3 blk(TB);
  auto b1d = [&](long long total) { return dim3((unsigned)((total + TB - 1) / TB)); };

  for (int g = 0; g < 4; ++g) {
    const float* x     = (const float*)d_in[g * 4 + 0];
    const int*   ei    = (const int*)d_in[g * 4 + 1];
    const int*   nidx  = ei;          // row 0: node indices
    const int*   eidx  = ei + E;      // row 1: hyperedge indices
    const int*   hnode = (const int*)d_in[g * 4 + 2];
    const float* hlen  = (const float*)d_in[g * 4 + 3];

    // degree vectors -> D^-1 (degn), B^-1 (dege)
    hipMemsetAsync(degn, 0, sizeof(float) * (size_t)N, stream);
    hipMemsetAsync(dege, 0, sizeof(float) * (size_t)N, stream);
    deg_count_kernel<<<b1d(E), blk, 0, stream>>>(nidx, eidx, degn, dege, E);
    invert_kernel<<<b1d(N), blk, 0, stream>>>(degn, N);
    invert_kernel<<<b1d(N), blk, 0, stream>>>(dege, N);

    // ---- conv1: 128 -> 256 ----
    gemm_wmma_f32<4><<<dim3(N / 16, D1 / 64), dim3(128), 0, stream>>>(x, W1, buf0, N, F_IN, D1);
    hipMemsetAsync(buf1, 0, sizeof(float) * (size_t)N * D1, stream);
    scatter_add_kernel<D1><<<b1d((long long)E * (D1 / 4)), blk, 0, stream>>>(
        buf0, nidx, eidx, nullptr, buf1, E);
    hipMemsetAsync(buf0, 0, sizeof(float) * (size_t)N * D1, stream);
    scatter_add_kernel<D1><<<b1d((long long)E * (D1 / 4)), blk, 0, stream>>>(
        buf1, eidx, nidx, dege, buf0, E);                         // B^-1 folded in
    scale_bias_relu_kernel<D1><<<b1d((long long)N * D1), blk, 0, stream>>>(
        buf0, degn, b1, buf0, (long long)N * D1);                 // h1 in buf0

    // ---- conv2: 256 -> 64 ----
    float* xw2 = buf1;                       // quarter 0 of buf1
    float* te2 = buf1 + (size_t)N * D2;      // quarter 1
    float* tn2 = buf1 + (size_t)2 * N * D2;  // quarter 2
    gemm_wmma_f32<4><<<dim3(N / 16, 1), dim3(128), 0, stream>>>(buf0, W2, xw2, N, D1, D2);
    hipMemsetAsync(te2, 0, sizeof(float) * (size_t)N * D2, stream);
    scatter_add_kernel<D2><<<b1d((long long)E * (D2 / 4)), blk, 0, stream>>>(
        xw2, nidx, eidx, nullptr, te2, E);
    hipMemsetAsync(tn2, 0, sizeof(float) * (size_t)N * D2, stream);
    scatter_add_kernel<D2><<<b1d((long long)E * (D2 / 4)), blk, 0, stream>>>(
        te2, eidx, nidx, dege, tn2, E);
    scale_bias_relu_kernel<D2><<<b1d((long long)N * D2), blk, 0, stream>>>(
        tn2, degn, b2, h2[g], (long long)N * D2);

    // ---- hyperedge embedding ----
    hye_emb_kernel<<<dim3((N + 3) / 4), dim3(256), 0, stream>>>(h2[g], hnode, hlen, he[g], N);
  }

  // ---- fusion attention ----
  Ptr4 zc = { h2[1], he[0], h2[2], he[2] };  // x1_mc, x2_cm, x1_cc, x2_cc
  Ptr4 zm = { h2[0], he[1], h2[3], he[3] };  // x1_cm, x2_mc, x1_mm, x2_mm
  hipMemsetAsync(scores, 0, sizeof(float) * 8, stream);
  dim3 agrid((N + TB - 1) / TB, 4);
  attn_score_kernel<<<agrid, blk, 0, stream>>>(zc, acW1, acb1, acW2, scores + 0, N);
  attn_score_kernel<<<agrid, blk, 0, stream>>>(zm, amW1, amb1, amW2, scores + 4, N);
  softmax_beta_kernel<<<1, 1, 0, stream>>>(scores, beta, 1.0f / (float)N);

  float* out = (float*)d_out;
  combine_kernel<<<b1d((long long)N * D2), blk, 0, stream>>>(zc, beta + 0, out, (long long)N * D2);
  combine_kernel<<<b1d((long long)N * D2), blk, 0, stream>>>(zm, beta + 4, out + (size_t)N * D2,
                                                             (long long)N * D2);
}